// VSGCLayer_26834955666033
// MI455X (gfx1250) — compile-verified
//
#include <hip/hip_runtime.h>
#include <hip/hip_bf16.h>

typedef float v2f __attribute__((ext_vector_type(2)));
typedef float v8f __attribute__((ext_vector_type(8)));

#define IDIM 128
#define ODIM 64
#define K_HOPS 4

// ---------------------------------------------------------------- utilities
__global__ void vsgc_zero(float* __restrict__ p, int n) {
  int i = blockIdx.x * blockDim.x + threadIdx.x;
  if (i < n) p[i] = 0.0f;
}

__global__ void vsgc_degree(const int* __restrict__ dst, float* __restrict__ deg, int E) {
  int e = blockIdx.x * blockDim.x + threadIdx.x;
  if (e < E) atomicAdd(&deg[dst[e]], 1.0f);
}

__global__ void vsgc_norm(const float* __restrict__ deg, float* __restrict__ nrm, int n) {
  int i = blockIdx.x * blockDim.x + threadIdx.x;
  if (i < n) nrm[i] = 1.0f / sqrtf(fmaxf(deg[i], 1.0f));
}

// ------------------------------------------------- H[N,64] = X[N,128] @ W^T
// One wave handles a 16-row strip; 4 tiles of 16x16 accumulated over K=128
// in steps of 4 via V_WMMA_F32_16X16X4_F32 (full fp32 precision).
__global__ void __launch_bounds__(128) vsgc_gemm_wmma(
    const float* __restrict__ X, const float* __restrict__ W,
    float* __restrict__ H, int N) {
  const int wave = blockIdx.x * (blockDim.x >> 5) + (threadIdx.x >> 5);
  const int lane = threadIdx.x & 31;
  const int m0   = wave * 16;
  if (m0 >= N) return;                 // wave-uniform: EXEC stays all-1s for WMMA

  const int half = lane >> 4;          // 0 -> K{0,1}, 1 -> K{2,3} per A-frag layout
  const int l16  = lane & 15;

  int row = m0 + l16; if (row >= N) row = N - 1;   // clamp (N is a multiple of 16 here)
  const float* __restrict__ a0 = X + (size_t)row * IDIM + 2 * half;
  const float* __restrict__ b0 = W + (size_t)l16 * IDIM + 2 * half;

  v8f acc[4];
#pragma unroll
  for (int t = 0; t < 4; ++t)
    acc[t] = (v8f){0.f, 0.f, 0.f, 0.f, 0.f, 0.f, 0.f, 0.f};

  for (int kb = 0; kb < IDIM; kb += 4) {
    v2f a; a.x = a0[kb]; a.y = a0[kb + 1];
#pragma unroll
    for (int t = 0; t < 4; ++t) {
      const float* __restrict__ bt = b0 + (size_t)t * 16 * IDIM;
      v2f b; b.x = bt[kb]; b.y = bt[kb + 1];
      // D = A(16x4) x B(4x16) + C ; 8 args: neg_a, A, neg_b, B, c_mod, C, reuse_a, reuse_b
      acc[t] = __builtin_amdgcn_wmma_f32_16x16x4_f32(
          false, a, false, b, (short)0, acc[t], false, false);
    }
  }

#pragma unroll
  for (int t = 0; t < 4; ++t) {
#pragma unroll
    for (int r = 0; r < 8; ++r) {
      int orow = m0 + r + 8 * half;    // C/D layout: VGPR r = rows r / r+8
      if (orow < N) H[(size_t)orow * ODIM + t * 16 + l16] = acc[t][r];
    }
  }
}

// ------------------------------------------------------------- ri = h*norm^2
__global__ void vsgc_ri(const float* __restrict__ cur, const float* __restrict__ nrm,
                        float* __restrict__ ri, int total) {
  int idx = blockIdx.x * blockDim.x + threadIdx.x;
  if (idx < total) {
    float nv = nrm[idx >> 6];
    ri[idx] = cur[idx] * nv * nv;
  }
}

// -------------------------------- gather(src)*norm[src] -> atomic scatter(dst)
// 16 lanes per edge, float4 per lane: 128-bit coalesced gather of the 256B row.
__global__ void vsgc_scatter(const float* __restrict__ cur, const float* __restrict__ nrm,
                             const int* __restrict__ src, const int* __restrict__ dst,
                             float* __restrict__ acc, int E) {
  int gid  = blockIdx.x * blockDim.x + threadIdx.x;
  int e    = gid >> 4;
  int lane = gid & 15;
  if (e >= E) return;
  int s = src[e], d = dst[e];
  if (s == d) return;                 // self-loops removed before propagation
  float ns = nrm[s];
  const float4* __restrict__ row = (const float4*)(cur + (size_t)s * ODIM);
  float4 v = row[lane];
  float* arow = acc + (size_t)d * ODIM + lane * 4;
  atomicAdd(arow + 0, v.x * ns);
  atomicAdd(arow + 1, v.y * ns);
  atomicAdd(arow + 2, v.z * ns);
  atomicAdd(arow + 3, v.w * ns);
}

// ---- h_new = a*(acc*norm) + a*ri + (1-a)*h_pre ; resets acc for next hop.
// cur and outp may alias (h_pre == h after every hop), so no __restrict there.
__global__ void vsgc_combine(float* __restrict__ acc, const float* __restrict__ ri,
                             const float* cur, const float* __restrict__ nrm,
                             float* outp, int total) {
  int idx = blockIdx.x * blockDim.x + threadIdx.x;
  if (idx >= total) return;
  float v = acc[idx] * nrm[idx >> 6];
  acc[idx] = 0.0f;                    // fused re-zero: saves a full pass per hop
  outp[idx] = 0.5f * v + 0.5f * ri[idx] + 0.5f * cur[idx];
}

static inline int gridFor(long long n, int b) { return (int)((n + b - 1) / b); }

extern "C" void kernel_launch(void* const* d_in, const int* in_sizes, int n_in,
                              void* d_out, int out_size, void* d_ws, size_t ws_size,
                              hipStream_t stream) {
  const float* features = (const float*)d_in[0];   // [N,128]
  const float* W        = (const float*)d_in[1];   // [64,128]
  const int*   src      = (const int*)d_in[2];     // [E]
  const int*   dst      = (const int*)d_in[3];     // [E]
  float*       out      = (float*)d_out;           // [N,64]

  const int N = in_sizes[0] / IDIM;
  const int E = in_sizes[2];
  const int total = N * ODIM;

  // workspace layout (floats), regions padded to 256-element alignment
  size_t NP = ((size_t)N + 255) & ~(size_t)255;
  float* ws  = (float*)d_ws;
  float* deg = ws;                 // [N]
  float* nrm = ws + NP;            // [N]
  float* cur = ws + 2 * NP;        // [N*64]  (doubles as h_pre)
  float* ri  = cur + (size_t)total;// [N*64]
  float* acc = ri + (size_t)total; // [N*64]

  const int B = 256;

  // degree + norm
  vsgc_zero<<<gridFor(N, B), B, 0, stream>>>(deg, N);
  vsgc_zero<<<gridFor(total, B), B, 0, stream>>>(acc, total);
  vsgc_degree<<<gridFor(E, B), B, 0, stream>>>(dst, deg, E);
  vsgc_norm<<<gridFor(N, B), B, 0, stream>>>(deg, nrm, N);

  // projection via fp32 WMMA: one wave per 16-row strip, 4 waves per block
  int strips = (N + 15) / 16;
  vsgc_gemm_wmma<<<gridFor(strips, 4), 128, 0, stream>>>(features, W, cur, N);

  // ri = h * norm^2
  vsgc_ri<<<gridFor(total, B), B, 0, stream>>>(cur, nrm, ri, total);

  // K hops of propagate + blend (last hop writes d_out)
  for (int hop = 0; hop < K_HOPS; ++hop) {
    vsgc_scatter<<<gridFor((long long)E * 16, B), B, 0, stream>>>(cur, nrm, src, dst, acc, E);
    float* dstbuf = (hop == K_HOPS - 1) ? out : cur;
    vsgc_combine<<<gridFor(total, B), B, 0, stream>>>(acc, ri, cur, nrm, dstbuf, total);
  }
}